// pi_GNN_78125455114293
// MI455X (gfx1250) — compile-verified
//
#include <hip/hip_runtime.h>
#include <cstdint>

#define BATCH 2048
#define NMAX  128
#define HN    64
#define HD    128
#define FD    16
#define OT_ITERS 100
#define ZSTR  133   // Sinkhorn LDS stride: odd -> conflict-free rows & columns
#define PSTR  130   // k_align LDS stride: even, 8B-aligned pairs, 2*l%64 distinct

typedef float v2f __attribute__((ext_vector_type(2)));
typedef float v8f __attribute__((ext_vector_type(8)));
typedef unsigned int u32x4 __attribute__((ext_vector_type(4)));
typedef int i32x4 __attribute__((ext_vector_type(4)));
typedef int i32x8 __attribute__((ext_vector_type(8)));

__device__ __forceinline__ float wave_max(float v) {
#pragma unroll
  for (int m = 16; m >= 1; m >>= 1) v = fmaxf(v, __shfl_xor(v, m, 32));
  return v;
}
__device__ __forceinline__ float wave_sum(float v) {
#pragma unroll
  for (int m = 16; m >= 1; m >>= 1) v += __shfl_xor(v, m, 32);
  return v;
}

// ---------------------------------------------------------------------------
// Tensor Data Mover: 2D f32 tile Global->LDS with hardware row padding.
// pad_interval_code: interval = 2^(code+1) DWORDs; pad_amount_code: code+1 DWORDs.
// ---------------------------------------------------------------------------
#if __has_builtin(__builtin_amdgcn_tensor_load_to_lds)
#define HAVE_TDM 1
__device__ __forceinline__ void tdm_load_tile_f32(
    void* lds_dst, const float* gsrc, int tile_w, int tile_h, int gstride_elems,
    int pad_interval_code, int pad_amount_code, int pad_en) {
  uint64_t ga = (uint64_t)(uintptr_t)gsrc;
  uint32_t lo = (uint32_t)(uintptr_t)lds_dst;  // generic LDS addr low 32 = LDS offset
  u32x4 g0;
  g0.x = 1u;                                           // count=1, user descriptor
  g0.y = lo;                                           // lds_addr
  g0.z = (uint32_t)ga;                                 // global_addr[31:0]
  g0.w = (uint32_t)((ga >> 32) & 0x01FFFFFFu) | (2u << 30);  // addr[56:32], type=2
  i32x8 g1;
  g1[0] = (2 << 16) | (pad_en << 20) | (pad_interval_code << 22) |
          (pad_amount_code << 25);                     // data_size=4B + padding
  g1[1] = (tile_w & 0xFFFF) << 16;                     // tensor_dim0 lo16 (bits 63:48)
  g1[2] = ((tile_w >> 16) & 0xFFFF) | ((tile_h & 0xFFFF) << 16);  // dim0 hi / dim1 lo
  g1[3] = ((tile_h >> 16) & 0xFFFF) | ((tile_w & 0xFFFF) << 16);  // dim1 hi / tile_dim0
  g1[4] = (tile_h & 0xFFFF);                           // tile_dim1 (tile_dim2=0)
  g1[5] = gstride_elems;                               // tensor_dim0_stride lo32
  g1[6] = 0;
  g1[7] = 0;
  i32x4 gz = {0, 0, 0, 0};
#if __clang_major__ >= 23
  i32x8 gz8 = {0, 0, 0, 0, 0, 0, 0, 0};
  __builtin_amdgcn_tensor_load_to_lds(g0, g1, gz, gz, gz8, 0);
#else
  __builtin_amdgcn_tensor_load_to_lds(g0, g1, gz, gz, 0);
#endif
}
#endif

// ---------------------------------------------------------------------------
// K1: struc/feats MLP + scores GEMM (WMMA f32 16x16x4)
// ---------------------------------------------------------------------------
__global__ __launch_bounds__(256) void k_scores(
    const float* __restrict__ x, const float* __restrict__ w1,
    const float* __restrict__ w2,
    const float* __restrict__ fc2w, const float* __restrict__ fc2b,
    const float* __restrict__ fc3w, const float* __restrict__ fc3b,
    float* __restrict__ scores1, float* __restrict__ scores2) {
  __shared__ float sT[HD * 129];  // transposed features: sT[c*129 + n]
  const int b    = blockIdx.x;
  const int tid  = threadIdx.x;
  const int lane = tid & 31;
  const int wave = tid >> 5;
  const int l    = lane & 15;
  const int koff = (lane >> 4) << 1;   // 0 or 2
  const int mhi  = (lane >> 4) * 8;

  for (int phase = 0; phase < 2; ++phase) {
    if (phase == 0) {
      for (int idx = tid; idx < NMAX * HD; idx += 256) {
        int n = idx >> 7, c = idx & (HD - 1);
        const float* xr = x + ((size_t)b * NMAX + n) * 18;
        float v = fmaf(xr[0], fc2w[2 * c], fmaf(xr[1], fc2w[2 * c + 1], fc2b[c]));
        sT[c * 129 + n] = fmaxf(v, 0.0f);
      }
    } else {
      for (int idx = tid; idx < NMAX * HD; idx += 256) {
        int n = idx >> 7, c = idx & (HD - 1);
        const float* xr = x + ((size_t)b * NMAX + n) * 18 + 2;
        const float* wr = fc3w + c * FD;
        float v = fc3b[c];
#pragma unroll
        for (int f = 0; f < FD; ++f) v = fmaf(xr[f], wr[f], v);
        sT[c * 129 + n] = fmaxf(v, 0.0f);
      }
    }
    __syncthreads();

    const float* W = phase ? w2 : w1;
    float* outp    = phase ? scores2 : scores1;
    for (int t = 0; t < 4; ++t) {        // 32 tiles / 8 waves
      int tile = wave * 4 + t;
      int mt = tile >> 3, nt = tile & 7;
      const float* Arow = W + (mt * 16 + l) * HD;
      v8f acc = {0.f, 0.f, 0.f, 0.f, 0.f, 0.f, 0.f, 0.f};
      for (int k0 = 0; k0 < HD; k0 += 4) {
        v2f a, bb;
        a.x  = Arow[k0 + koff];
        a.y  = Arow[k0 + koff + 1];
        bb.x = sT[(k0 + koff) * 129 + nt * 16 + l];
        bb.y = sT[(k0 + koff + 1) * 129 + nt * 16 + l];
        acc = __builtin_amdgcn_wmma_f32_16x16x4_f32(false, a, false, bb,
                                                    (short)0, acc, false, false);
      }
#pragma unroll
      for (int r = 0; r < 8; ++r) {
        int M  = mt * 16 + r + mhi;
        int Nn = nt * 16 + l;
        outp[((size_t)b * HN + M) * NMAX + Nn] = fmaxf(acc[r], 0.0f);
      }
    }
    __syncthreads();
  }
}

// ---------------------------------------------------------------------------
// K2: fused double Sinkhorn (100 iters, both OTs LDS-resident); TDM-staged Z.
// ---------------------------------------------------------------------------
__global__ __launch_bounds__(256) void k_sinkhorn(
    const float* __restrict__ scores2, const float* __restrict__ alpha_p,
    const float* __restrict__ bin_p, float* __restrict__ pm_io) {
  __shared__ float Z1[65 * ZSTR];
  __shared__ float Z2[65 * ZSTR];
  __shared__ float u1[65], u2[65];
  __shared__ float v1[129], v2[129];
  const int b    = blockIdx.x;
  const int tid  = threadIdx.x;
  const int lane = tid & 31;
  const int wave = tid >> 5;
  const float binv    = bin_p[0];
  const float norm    = -__logf(192.0f);
  const float lmu_bin = __logf(128.0f) + norm;
  const float lnu_bin = __logf(64.0f) + norm;

#ifdef HAVE_TDM
  // TDM: 64x128 f32 tiles, pad 5 DWORDs every 128 -> LDS row stride 133
  if (wave == 0) {
    tdm_load_tile_f32(Z1, pm_io + (size_t)b * HN * NMAX, NMAX, HN, NMAX, 6, 4, 1);
    tdm_load_tile_f32(Z2, scores2 + (size_t)b * HN * NMAX, NMAX, HN, NMAX, 6, 4, 1);
  }
  // dustbin row (i=64, all j) and dustbin column (j=128, i<64)
  for (int idx = tid; idx < 129 + HN; idx += 256) {
    if (idx < 129) { Z1[HN * ZSTR + idx] = binv; Z2[HN * ZSTR + idx] = binv; }
    else {
      int i = idx - 129;
      Z1[i * ZSTR + NMAX] = binv; Z2[i * ZSTR + NMAX] = binv;
    }
  }
  if (wave == 0) __builtin_amdgcn_s_wait_tensorcnt(0);
#else
  for (int idx = tid; idx < 65 * 129; idx += 256) {
    int i = idx / 129, j = idx - i * 129;
    float z1, z2;
    if (i < HN && j < NMAX) {
      z1 = pm_io[((size_t)b * HN + i) * NMAX + j];
      z2 = scores2[((size_t)b * HN + i) * NMAX + j];
    } else { z1 = binv; z2 = binv; }
    Z1[i * ZSTR + j] = z1;
    Z2[i * ZSTR + j] = z2;
  }
#endif
  for (int idx = tid; idx < 129; idx += 256) {
    v1[idx] = 0.f; v2[idx] = 0.f;
    if (idx < 65) { u1[idx] = 0.f; u2[idx] = 0.f; }
  }
  __syncthreads();

  for (int it = 0; it < OT_ITERS; ++it) {
    // u-update: row logsumexp over 129 columns
    for (int i = wave; i < 65; i += 8) {
      float m1 = -__builtin_inff(), m2 = -__builtin_inff();
      for (int j = lane; j < 129; j += 32) {
        m1 = fmaxf(m1, Z1[i * ZSTR + j] + v1[j]);
        m2 = fmaxf(m2, Z2[i * ZSTR + j] + v2[j]);
      }
      m1 = wave_max(m1); m2 = wave_max(m2);
      float s1 = 0.f, s2 = 0.f;
      for (int j = lane; j < 129; j += 32) {
        s1 += __expf(Z1[i * ZSTR + j] + v1[j] - m1);
        s2 += __expf(Z2[i * ZSTR + j] + v2[j] - m2);
      }
      s1 = wave_sum(s1); s2 = wave_sum(s2);
      if (lane == 0) {
        float lm = (i < HN) ? norm : lmu_bin;
        u1[i] = lm - (m1 + __logf(s1));
        u2[i] = lm - (m2 + __logf(s2));
      }
    }
    __syncthreads();
    // v-update: column logsumexp over 65 rows
    for (int j = wave; j < 129; j += 8) {
      float m1 = -__builtin_inff(), m2 = -__builtin_inff();
      for (int i = lane; i < 65; i += 32) {
        m1 = fmaxf(m1, Z1[i * ZSTR + j] + u1[i]);
        m2 = fmaxf(m2, Z2[i * ZSTR + j] + u2[i]);
      }
      m1 = wave_max(m1); m2 = wave_max(m2);
      float s1 = 0.f, s2 = 0.f;
      for (int i = lane; i < 65; i += 32) {
        s1 += __expf(Z1[i * ZSTR + j] + u1[i] - m1);
        s2 += __expf(Z2[i * ZSTR + j] + u2[i] - m2);
      }
      s1 = wave_sum(s1); s2 = wave_sum(s2);
      if (lane == 0) {
        float ln = (j < NMAX) ? norm : lnu_bin;
        v1[j] = ln - (m1 + __logf(s1));
        v2[j] = ln - (m2 + __logf(s2));
      }
    }
    __syncthreads();
  }

  const float aw = 1.0f / (1.0f + __expf(-alpha_p[0]));
  for (int idx = tid; idx < HN * NMAX; idx += 256) {
    int i = idx >> 7, j = idx & (NMAX - 1);
    float p1 = __expf(Z1[i * ZSTR + j] + u1[i] + v1[j] - norm);
    float p2 = __expf(Z2[i * ZSTR + j] + u2[i] + v2[j] - norm);
    pm_io[((size_t)b * HN + i) * NMAX + j] = aw * p1 + (1.0f - aw) * p2;
  }
}

// ---------------------------------------------------------------------------
// K3: three WMMA f32 GEMMs with Pm (TDM-staged, stride 130 for b64 fragments)
// ---------------------------------------------------------------------------
__global__ __launch_bounds__(256) void k_align(
    const float* __restrict__ Pm, const float* __restrict__ adj,
    const float* __restrict__ x,
    float* __restrict__ adj2, float* __restrict__ featsal) {
  __shared__ float Ps[HN * PSTR];
  __shared__ float Aal[HN * PSTR];
  const int b    = blockIdx.x;
  const int tid  = threadIdx.x;
  const int lane = tid & 31;
  const int wave = tid >> 5;
  const int l    = lane & 15;
  const int koff = (lane >> 4) << 1;
  const int mhi  = (lane >> 4) * 8;

#ifdef HAVE_TDM
  // pad 2 DWORDs every 128 -> LDS row stride 130
  if (wave == 0) {
    tdm_load_tile_f32(Ps, Pm + (size_t)b * HN * NMAX, NMAX, HN, NMAX, 6, 1, 1);
    __builtin_amdgcn_s_wait_tensorcnt(0);
  }
#else
  for (int idx = tid; idx < HN * NMAX; idx += 256) {
    int i = idx >> 7, j = idx & (NMAX - 1);
    Ps[i * PSTR + j] = Pm[((size_t)b * HN + i) * NMAX + j];
  }
#endif
  __syncthreads();

  // GEMM1: Aal[m][d] = sum_n Pm[m][n] * adj[b][n][d]
  for (int t = 0; t < 4; ++t) {
    int tile = wave * 4 + t;
    int mt = tile >> 3, nt = tile & 7;
    const float* arow = &Ps[(mt * 16 + l) * PSTR];
    v8f acc = {0.f, 0.f, 0.f, 0.f, 0.f, 0.f, 0.f, 0.f};
    for (int k0 = 0; k0 < NMAX; k0 += 4) {
      v2f a, bb;
      a.x = arow[k0 + koff];
      a.y = arow[k0 + koff + 1];
      const float* brow = adj + ((size_t)b * NMAX + (k0 + koff)) * NMAX + nt * 16 + l;
      bb.x = brow[0];
      bb.y = brow[NMAX];
      acc = __builtin_amdgcn_wmma_f32_16x16x4_f32(false, a, false, bb,
                                                  (short)0, acc, false, false);
    }
#pragma unroll
    for (int r = 0; r < 8; ++r)
      Aal[(mt * 16 + r + mhi) * PSTR + nt * 16 + l] = acc[r];
  }
  __syncthreads();

  // GEMM2: adj2[m][k'] = sum_d Aal[m][d] * Pm[k'][d]
  for (int t = 0; t < 2; ++t) {
    int tile = wave * 2 + t;
    int mt = tile >> 2, nt = tile & 3;
    const float* arow = &Aal[(mt * 16 + l) * PSTR];
    const float* bcol = &Ps[(nt * 16 + l) * PSTR];
    v8f acc = {0.f, 0.f, 0.f, 0.f, 0.f, 0.f, 0.f, 0.f};
    for (int k0 = 0; k0 < NMAX; k0 += 4) {
      v2f a, bb;
      a.x  = arow[k0 + koff];
      a.y  = arow[k0 + koff + 1];
      bb.x = bcol[k0 + koff];
      bb.y = bcol[k0 + koff + 1];
      acc = __builtin_amdgcn_wmma_f32_16x16x4_f32(false, a, false, bb,
                                                  (short)0, acc, false, false);
    }
#pragma unroll
    for (int r = 0; r < 8; ++r)
      adj2[(size_t)b * (HN * HN) + (mt * 16 + r + mhi) * HN + nt * 16 + l] = acc[r];
  }

  // GEMM3: featsal[m][f] = sum_n Pm[m][n] * x[b,n,2+f]
  if (wave < 4) {
    int mt = wave;
    const float* arow = &Ps[(mt * 16 + l) * PSTR];
    v8f acc = {0.f, 0.f, 0.f, 0.f, 0.f, 0.f, 0.f, 0.f};
    for (int k0 = 0; k0 < NMAX; k0 += 4) {
      v2f a, bb;
      a.x = arow[k0 + koff];
      a.y = arow[k0 + koff + 1];
      const float* xr = x + ((size_t)b * NMAX + (k0 + koff)) * 18 + 2 + l;
      bb.x = xr[0];
      bb.y = xr[18];
      acc = __builtin_amdgcn_wmma_f32_16x16x4_f32(false, a, false, bb,
                                                  (short)0, acc, false, false);
    }
#pragma unroll
    for (int r = 0; r < 8; ++r)
      featsal[(size_t)b * (HN * FD) + (mt * 16 + r + mhi) * FD + l] = acc[r];
  }
}

// ---------------------------------------------------------------------------
// weight transpose so FC dots read weights coalesced (L2-resident)
// ---------------------------------------------------------------------------
__global__ __launch_bounds__(256) void k_transpose(const float* __restrict__ in,
                                                   float* __restrict__ outp,
                                                   int rows, int cols) {
  int total = rows * cols;
  for (int idx = blockIdx.x * blockDim.x + threadIdx.x; idx < total;
       idx += gridDim.x * blockDim.x) {
    int r = idx / cols, c = idx - r * cols;
    outp[c * rows + r] = in[idx];
  }
}

// ---------------------------------------------------------------------------
// K4: per-sample LayerNorm + fc4/fc5 + fc6/fc7 + log_softmax
// ---------------------------------------------------------------------------
__global__ __launch_bounds__(256) void k_tail(
    const float* __restrict__ adj2, const float* __restrict__ featsal,
    const float* __restrict__ ln1g, const float* __restrict__ ln1b,
    const float* __restrict__ fc4wT, const float* __restrict__ fc4b,
    const float* __restrict__ ln2g, const float* __restrict__ ln2b,
    const float* __restrict__ fc5wT, const float* __restrict__ fc5b,
    const float* __restrict__ fc6w, const float* __restrict__ fc6b,
    const float* __restrict__ fc7w, const float* __restrict__ fc7b,
    float* __restrict__ outp) {
  __shared__ float sv[4096];
  __shared__ float so[512];
  __shared__ float sred[16];
  __shared__ float smisc[4];
  const int b    = blockIdx.x;
  const int tid  = threadIdx.x;
  const int lane = tid & 31;
  const int wave = tid >> 5;

  // ---- adj branch: LN(4096) + fc4 ----
  {
    const float* row = adj2 + (size_t)b * 4096;
    float s = 0.f, s2 = 0.f;
    for (int k = tid; k < 4096; k += 256) {
      float v = row[k]; sv[k] = v; s += v; s2 = fmaf(v, v, s2);
    }
    s = wave_sum(s); s2 = wave_sum(s2);
    if (lane == 0) { sred[wave] = s; sred[8 + wave] = s2; }
    __syncthreads();
    if (tid == 0) {
      float ts = 0.f, ts2 = 0.f;
      for (int w = 0; w < 8; ++w) { ts += sred[w]; ts2 += sred[8 + w]; }
      float mean = ts * (1.0f / 4096.0f);
      float var  = ts2 * (1.0f / 4096.0f) - mean * mean;
      smisc[0] = mean; smisc[1] = rsqrtf(var + 1e-5f);
    }
    __syncthreads();
    float mean = smisc[0], rstd = smisc[1];
    for (int k = tid; k < 4096; k += 256)
      sv[k] = (sv[k] - mean) * rstd * ln1g[k] + ln1b[k];
    __syncthreads();
    __builtin_prefetch(&fc4wT[tid], 0, 1);
    float acc = fc4b[tid];
    for (int k = 0; k < 4096; ++k) acc = fmaf(sv[k], fc4wT[k * 256 + tid], acc);
    so[tid] = fmaxf(acc, 0.0f);
  }
  __syncthreads();

  // ---- feats branch: LN(1024) + fc5 ----
  {
    const float* row = featsal + (size_t)b * 1024;
    float s = 0.f, s2 = 0.f;
    for (int k = tid; k < 1024; k += 256) {
      float v = row[k]; sv[k] = v; s += v; s2 = fmaf(v, v, s2);
    }
    s = wave_sum(s); s2 = wave_sum(s2);
    if (lane == 0) { sred[wave] = s; sred[8 + wave] = s2; }
    __syncthreads();
    if (tid == 0) {
      float ts = 0.f, ts2 = 0.f;
      for (int w = 0; w < 8; ++w) { ts += sred[w]; ts2 += sred[8 + w]; }
      float mean = ts * (1.0f / 1024.0f);
      float var  = ts2 * (1.0f / 1024.0f) - mean * mean;
      smisc[0] = mean; smisc[1] = rsqrtf(var + 1e-5f);
    }
    __syncthreads();
    float mean = smisc[0], rstd = smisc[1];
    for (int k = tid; k < 1024; k += 256)
      sv[k] = (sv[k] - mean) * rstd * ln2g[k] + ln2b[k];
    __syncthreads();
    float acc = fc5b[tid];
    for (int k = 0; k < 1024; ++k) acc = fmaf(sv[k], fc5wT[k * 256 + tid], acc);
    so[256 + tid] = fmaxf(acc, 0.0f);
  }
  __syncthreads();

  // fc6: 512 -> 64, relu
  if (tid < 64) {
    float acc = fc6b[tid];
    const float* wr = fc6w + tid * 512;
    for (int k = 0; k < 512; ++k) acc = fmaf(so[k], wr[k], acc);
    sv[tid] = fmaxf(acc, 0.0f);
  }
  __syncthreads();
  // fc7: 64 -> 10
  if (tid < 10) {
    float acc = fc7b[tid];
    const float* wr = fc7w + tid * 64;
    for (int k = 0; k < 64; ++k) acc = fmaf(sv[k], wr[k], acc);
    sv[256 + tid] = acc;
  }
  __syncthreads();
  if (tid == 0) {
    float m = -__builtin_inff();
    for (int o = 0; o < 10; ++o) m = fmaxf(m, sv[256 + o]);
    float s = 0.f;
    for (int o = 0; o < 10; ++o) s += __expf(sv[256 + o] - m);
    smisc[0] = m + __logf(s);
  }
  __syncthreads();
  if (tid < 10) outp[(size_t)b * 10 + tid] = sv[256 + tid] - smisc[0];
}

// ---------------------------------------------------------------------------
extern "C" void kernel_launch(void* const* d_in, const int* in_sizes, int n_in,
                              void* d_out, int out_size, void* d_ws, size_t ws_size,
                              hipStream_t stream) {
  const float* x     = (const float*)d_in[0];
  const float* adj   = (const float*)d_in[1];
  const float* w1    = (const float*)d_in[2];
  const float* w2    = (const float*)d_in[3];
  const float* alpha = (const float*)d_in[4];
  const float* binsc = (const float*)d_in[5];
  const float* fc2w  = (const float*)d_in[6];
  const float* fc2b  = (const float*)d_in[7];
  const float* fc3w  = (const float*)d_in[8];
  const float* fc3b  = (const float*)d_in[9];
  const float* ln1g  = (const float*)d_in[10];
  const float* ln1b  = (const float*)d_in[11];
  const float* fc4w  = (const float*)d_in[12];
  const float* fc4b  = (const float*)d_in[13];
  const float* ln2g  = (const float*)d_in[14];
  const float* ln2b  = (const float*)d_in[15];
  const float* fc5w  = (const float*)d_in[16];
  const float* fc5b  = (const float*)d_in[17];
  const float* fc6w  = (const float*)d_in[18];
  const float* fc6b  = (const float*)d_in[19];
  const float* fc7w  = (const float*)d_in[20];
  const float* fc7b  = (const float*)d_in[21];
  float* outp = (float*)d_out;
  (void)in_sizes; (void)n_in; (void)out_size; (void)ws_size;

  float* ws = (float*)d_ws;
  size_t off = 0;
  float* scores1 = ws + off; off += (size_t)BATCH * HN * NMAX;  // becomes Pm
  float* scores2 = ws + off; off += (size_t)BATCH * HN * NMAX;
  float* adj2    = ws + off; off += (size_t)BATCH * HN * HN;
  float* featsal = ws + off; off += (size_t)BATCH * HN * FD;
  float* fc4wT   = ws + off; off += (size_t)4096 * 256;
  float* fc5wT   = ws + off; off += (size_t)1024 * 256;

  k_scores<<<BATCH, 256, 0, stream>>>(x, w1, w2, fc2w, fc2b, fc3w, fc3b,
                                      scores1, scores2);
  k_transpose<<<1024, 256, 0, stream>>>(fc4w, fc4wT, 256, 4096);
  k_transpose<<<256, 256, 0, stream>>>(fc5w, fc5wT, 256, 1024);
  k_sinkhorn<<<BATCH, 256, 0, stream>>>(scores2, alpha, binsc, scores1);
  k_align<<<BATCH, 256, 0, stream>>>(scores1, adj, x, adj2, featsal);
  k_tail<<<BATCH, 256, 0, stream>>>(adj2, featsal, ln1g, ln1b, fc4wT, fc4b,
                                    ln2g, ln2b, fc5wT, fc5b, fc6w, fc6b,
                                    fc7w, fc7b, outp);
}